// VQ_86603720556686
// MI455X (gfx1250) — compile-verified
//
#include <hip/hip_runtime.h>
#include <hip/hip_bf16.h>

// CDNA5 / gfx1250, wave32. Uses V_WMMA_F32_16X16X4_F32 for the fp32 distance GEMM.

typedef __attribute__((ext_vector_type(2))) float v2f;
typedef __attribute__((ext_vector_type(8))) float v8f;

#define K_CODES    512
#define DIM        64
#define CB_STRIDE  68          // LDS row stride (floats): conflict-free ds_load_b64
#define HW         4096        // 64*64
#define CHW        262144      // 64*64*64
#define WAVES_PER_WG 8
#define TILE_ROWS  16
#define NUM_TILES  8192        // 131072 points / 16
#define TOTAL_ELEMS 8388608.0f // 32*64*64*64

__global__ void vq_init_out(float* out) {
    if (threadIdx.x == 0 && blockIdx.x == 0) out[0] = 0.0f;
}

__global__ void __launch_bounds__(256, 1)
vq_kernel(const float* __restrict__ in, const float* __restrict__ cb,
          float* __restrict__ out)
{
    extern __shared__ float smem[];
    float* lds_cb  = smem;                           // 512 * 68 floats
    float* lds_c2  = smem + K_CODES * CB_STRIDE;     // 512 floats
    int*   lds_idx = (int*)(lds_c2 + K_CODES);       // 8 waves * 16 rows

    const int tid = threadIdx.x;

    // ---- Stage codebook into LDS (float4 coalesced), padded rows ----
    for (int j = tid; j < (K_CODES * DIM) / 4; j += 256) {
        float4 v = ((const float4*)cb)[j];
        int k = j >> 4;            // (j*4)/64
        int c = (j & 15) << 2;     // (j*4)%64
        *(float4*)&lds_cb[k * CB_STRIDE + c] = v;
    }
    __syncthreads();

    // ---- Codebook squared norms ----
    for (int k = tid; k < K_CODES; k += 256) {
        const float* row = &lds_cb[k * CB_STRIDE];
        float s = 0.0f;
        #pragma unroll
        for (int d = 0; d < DIM; ++d) s += row[d] * row[d];
        lds_c2[k] = s;
    }
    __syncthreads();

    const int wave = tid >> 5;
    const int lane = tid & 31;
    const int n16  = lane & 15;    // A: row m; B: code-in-chunk n; C: column n
    const int h    = lane >> 4;    // half (K sub-stripe)

    const int tile = blockIdx.x * WAVES_PER_WG + wave;   // 0..8191
    const int p0   = tile * TILE_ROWS;
    const int img  = p0 >> 12;                           // /4096 (tiles never cross images)
    const int hw0  = p0 & 4095;
    const float* ibase = in + (size_t)img * CHW + hw0;   // element [d][m] at + d*4096 + m

    // ---- Load A fragments: a[s] = flat[m][4s+2h], flat[m][4s+2h+1] ----
    v2f a[16];
    #pragma unroll
    for (int s = 0; s < 16; ++s) {
        const int d0 = 4 * s + 2 * h;
        a[s].x = ibase[(size_t)d0 * HW + n16];
        a[s].y = ibase[(size_t)(d0 + 1) * HW + n16];
    }

    float minval[8];
    int   minidx[8];
    #pragma unroll
    for (int r = 0; r < 8; ++r) { minval[r] = 3.4e38f; minidx[r] = 0; }

    // B fragment base: codebook[k0+n][d0+2h .. +1] in padded LDS
    const float* bbase = lds_cb + n16 * CB_STRIDE + 2 * h;

    for (int c = 0; c < 32; ++c) {             // 32 chunks of 16 codes
        v8f acc = {};
        const float* bp = bbase + c * (16 * CB_STRIDE);
        #pragma unroll
        for (int s = 0; s < 16; ++s) {         // accumulate over D in K=4 steps
            v2f b = *(const v2f*)(bp + 4 * s); // ds_load_b64, conflict-free
            acc = __builtin_amdgcn_wmma_f32_16x16x4_f32(
                      false, a[s], false, b, (short)0, acc, false, false);
        }
        const int kk = (c << 4) + n16;
        const float c2 = lds_c2[kk];
        #pragma unroll
        for (int r = 0; r < 8; ++r) {          // d2 = c2 - 2*dot (+x2 is row-const)
            const float d2 = __builtin_fmaf(-2.0f, acc[r], c2);
            if (d2 < minval[r]) { minval[r] = d2; minidx[r] = kk; }
        }
    }

    // ---- Cross-lane argmin within each 16-lane half (first-index ties) ----
    #pragma unroll
    for (int msk = 1; msk <= 8; msk <<= 1) {
        #pragma unroll
        for (int r = 0; r < 8; ++r) {
            const float ov = __shfl_xor(minval[r], msk, 32);
            const int   oi = __shfl_xor(minidx[r], msk, 32);
            if (ov < minval[r] || (ov == minval[r] && oi < minidx[r])) {
                minval[r] = ov; minidx[r] = oi;
            }
        }
    }
    if (n16 == 0) {  // lanes 0 (rows 0..7) and 16 (rows 8..15)
        #pragma unroll
        for (int r = 0; r < 8; ++r) lds_idx[wave * 16 + 8 * h + r] = minidx[r];
    }
    __syncthreads();

    // ---- Gather quantized rows from LDS, write NCHW, accumulate loss ----
    const int m     = n16;
    const int bestk = lds_idx[wave * 16 + m];
    const float* qrow = lds_cb + bestk * CB_STRIDE;
    float* obase = out + 1 + (size_t)img * CHW + hw0 + m;
    float psum = 0.0f;
    #pragma unroll
    for (int s = 0; s < 16; ++s) {
        const int d0 = 4 * s + 2 * h;
        v2f q = *(const v2f*)(qrow + d0);      // 8B-aligned ds_load_b64
        const float dx = q.x - a[s].x;         // reuse A regs: no input re-read
        const float dy = q.y - a[s].y;
        psum += dx * dx + dy * dy;
        obase[(size_t)d0 * HW]       = q.x;    // 16-lane contiguous 64B segments
        obase[(size_t)(d0 + 1) * HW] = q.y;
    }

    // wave-wide reduction of squared error
    #pragma unroll
    for (int msk = 16; msk >= 1; msk >>= 1) psum += __shfl_xor(psum, msk, 32);
    if (lane == 0) atomicAdd(out, psum * (1.25f / TOTAL_ELEMS));
}

extern "C" void kernel_launch(void* const* d_in, const int* in_sizes, int n_in,
                              void* d_out, int out_size, void* d_ws, size_t ws_size,
                              hipStream_t stream) {
    const float* in  = (const float*)d_in[0];   // [32, 64, 64, 64] f32
    const float* cb  = (const float*)d_in[1];   // [512, 64] f32
    float* out = (float*)d_out;                 // [0]=c_loss, [1..]=quantized_st (NCHW)

    vq_init_out<<<1, 64, 0, stream>>>(out);

    const size_t shmem = (size_t)(K_CODES * CB_STRIDE + K_CODES) * sizeof(float)
                       + (size_t)(WAVES_PER_WG * 16) * sizeof(int);  // ~141.8 KB (<320KB/WGP)
    vq_kernel<<<NUM_TILES / WAVES_PER_WG, 256, shmem, stream>>>(in, cb, out);
}